// SAGEConv_81131932221713
// MI455X (gfx1250) — compile-verified
//
#include <hip/hip_runtime.h>
#include <hip/hip_bf16.h>

// ---------------------------------------------------------------------------
// GraphSAGE conv for MI455X (gfx1250, wave32).
//   support = scatter_add(h[edge_col] * edge_val, edge_row)      (L2-resident atomics)
//   x = LayerNorm(ReLU(concat(h @ Wself^T + bself, support @ Wneigh^T + bneigh)))
// GEMMs use V_WMMA_F32_16X16X4_F32 (full f32 precision matrix pipe).
// ---------------------------------------------------------------------------

typedef __attribute__((ext_vector_type(2))) float v2f;
typedef __attribute__((ext_vector_type(8))) float v8f;

#define DIN   128
#define DOUT  128
#define DCAT  256
#define TILE_M 16

// ---------------------------------------------------------------------------
// Kernel 1: zero the support accumulator (float4 stores).
// ---------------------------------------------------------------------------
__global__ __launch_bounds__(256) void sage_zero_kernel(float4* __restrict__ p, int n4) {
    int i = blockIdx.x * blockDim.x + threadIdx.x;
    if (i < n4) {
        p[i] = make_float4(0.f, 0.f, 0.f, 0.f);
    }
}

// ---------------------------------------------------------------------------
// Kernel 2: edge scatter (SpMM). One wave32 per edge; lane l owns the
// float4 chunk [4l, 4l+3] of the 128-wide feature row.
// global_load_b128 gather + 4x global_atomic_add_f32 (no-return) scatter.
// ---------------------------------------------------------------------------
__global__ __launch_bounds__(256) void sage_spmm_scatter(
    const float* __restrict__ h,
    const int*   __restrict__ erow,
    const int*   __restrict__ ecol,
    const float* __restrict__ eval,
    float*       __restrict__ support,
    int nEdges)
{
    const int lane = threadIdx.x & 31;
    const int wave = (blockIdx.x * blockDim.x + threadIdx.x) >> 5;
    if (wave >= nEdges) return;

    const int   r = erow[wave];
    const int   c = ecol[wave];
    const float v = eval[wave];

    const float4 src = ((const float4*)(h + (size_t)c * DIN))[lane];
    float* dst = support + (size_t)r * DIN + lane * 4;
    atomicAdd(dst + 0, src.x * v);
    atomicAdd(dst + 1, src.y * v);
    atomicAdd(dst + 2, src.z * v);
    atomicAdd(dst + 3, src.w * v);
}

// ---------------------------------------------------------------------------
// Kernel 3: fused dual GEMM (WMMA f32 16x16x4) + bias + ReLU + concat +
// LayerNorm(256). Block = 256 threads = 8 waves, owns 16 node rows.
// Wave w computes output columns [16w, 16w+16) of BOTH GEMMs.
// ---------------------------------------------------------------------------
__global__ __launch_bounds__(256) void sage_gemm_ln(
    const float* __restrict__ h,
    const float* __restrict__ support,
    const float* __restrict__ Wself,
    const float* __restrict__ bself,
    const float* __restrict__ Wneigh,
    const float* __restrict__ bneigh,
    const float* __restrict__ gamma,
    const float* __restrict__ beta,
    float*       __restrict__ out)
{
    __shared__ float lds_h[TILE_M * DIN];   // 8 KB
    __shared__ float lds_s[TILE_M * DIN];   // 8 KB
    __shared__ float lds_x[TILE_M * DCAT];  // 16 KB post-ReLU concat tile

    const int n0   = blockIdx.x * TILE_M;
    const int tid  = threadIdx.x;
    const int wave = tid >> 5;
    const int lane = tid & 31;

    // Cooperative float4 staging of the 16x128 h / support tiles.
    for (int i = tid; i < TILE_M * DIN / 4; i += 256) {
        const int r  = i >> 5;        // 32 float4 per row
        const int c4 = i & 31;
        ((float4*)lds_h)[i] = ((const float4*)(h       + (size_t)(n0 + r) * DIN))[c4];
        ((float4*)lds_s)[i] = ((const float4*)(support + (size_t)(n0 + r) * DIN))[c4];
    }
    __syncthreads();

    const int j0 = wave * 16;            // column tile within each GEMM
    // A-frag (16x4 f32): lanes 0-15 -> K = {0,1}; lanes 16-31 -> K = {2,3}
    const int mA     = lane & 15;
    const int kApair = (lane >> 4) * 2;
    // B-frag (4x16 f32): VGPR v, lanes 0-15 -> K = v; lanes 16-31 -> K = v+2
    const int nB     = lane & 15;
    const int kBoff  = (lane >> 4) * 2;

    #pragma unroll
    for (int g = 0; g < 2; ++g) {
        const float* Atile = g ? lds_s : lds_h;
        const float* W     = g ? Wneigh : Wself;
        const float* bias  = g ? bneigh : bself;

        v8f acc = {};
        #pragma unroll
        for (int k0 = 0; k0 < DIN; k0 += 4) {
            // ds_load_b64: two consecutive K values for this lane's A rows
            v2f a = *(const v2f*)(&Atile[mA * DIN + k0 + kApair]);
            // global_load_b64 from weights (hot in L2; W[j][k], B[k][n] = W[j0+n][k0+k])
            v2f b = *(const v2f*)(&W[(size_t)(j0 + nB) * DIN + k0 + kBoff]);
            acc = __builtin_amdgcn_wmma_f32_16x16x4_f32(
                /*neg_a=*/false, a, /*neg_b=*/false, b,
                /*c_mod=*/(short)0, acc, /*reuse_a=*/false, /*reuse_b=*/false);
        }

        // Bias + ReLU, write tile into the concat staging buffer.
        // C/D layout: VGPR r -> (M = r + 8*(lane>=16), N = lane&15)
        const float bv      = bias[j0 + (lane & 15)];
        const int   rowoff  = (lane >> 4) * 8;
        const int   colbase = g * DOUT + j0 + (lane & 15);
        #pragma unroll
        for (int r = 0; r < 8; ++r) {
            float v = acc[r] + bv;
            v = v > 0.f ? v : 0.f;
            lds_x[(r + rowoff) * DCAT + colbase] = v;
        }
    }
    __syncthreads();

    // LayerNorm over 256: wave w normalizes rows 2w and 2w+1.
    #pragma unroll
    for (int rr = 0; rr < 2; ++rr) {
        const int row = wave * 2 + rr;
        float vals[8];
        float s = 0.f, s2 = 0.f;
        #pragma unroll
        for (int i = 0; i < 8; ++i) {
            const float v = lds_x[row * DCAT + lane + 32 * i];
            vals[i] = v;
            s  += v;
            s2 += v * v;
        }
        #pragma unroll
        for (int m = 16; m >= 1; m >>= 1) {
            s  += __shfl_xor(s,  m, 32);
            s2 += __shfl_xor(s2, m, 32);
        }
        const float mean = s  * (1.f / 256.f);
        const float var  = s2 * (1.f / 256.f) - mean * mean;
        const float inv  = rsqrtf(var + 1e-5f);

        float* dst = out + (size_t)(n0 + row) * DCAT;
        #pragma unroll
        for (int i = 0; i < 8; ++i) {
            const int col = lane + 32 * i;
            dst[col] = (vals[i] - mean) * inv * gamma[col] + beta[col];
        }
    }
}

// ---------------------------------------------------------------------------
// Launcher
// ---------------------------------------------------------------------------
extern "C" void kernel_launch(void* const* d_in, const int* in_sizes, int n_in,
                              void* d_out, int out_size, void* d_ws, size_t ws_size,
                              hipStream_t stream) {
    const float* h      = (const float*)d_in[0];
    const int*   erow   = (const int*)  d_in[1];
    const int*   ecol   = (const int*)  d_in[2];
    const float* eval   = (const float*)d_in[3];
    const float* Wself  = (const float*)d_in[4];
    const float* bself  = (const float*)d_in[5];
    const float* Wneigh = (const float*)d_in[6];
    const float* bneigh = (const float*)d_in[7];
    const float* gamma  = (const float*)d_in[8];
    const float* beta   = (const float*)d_in[9];

    const int nNodes = in_sizes[0] / DIN;
    const int nEdges = in_sizes[1];

    float* support = (float*)d_ws;  // nNodes * 128 f32 = 25.6 MB

    // 1) zero the scatter accumulator
    {
        const int n4 = nNodes * DIN / 4;
        const int blocks = (n4 + 255) / 256;
        sage_zero_kernel<<<blocks, 256, 0, stream>>>((float4*)support, n4);
    }

    // 2) edge scatter: one wave32 per edge
    {
        const long long threads = (long long)nEdges * 32;
        const int blocks = (int)((threads + 255) / 256);
        sage_spmm_scatter<<<blocks, 256, 0, stream>>>(h, erow, ecol, eval, support, nEdges);
    }

    // 3) fused WMMA GEMMs + bias/ReLU/concat + LayerNorm (nNodes % 16 == 0)
    {
        const int blocks = nNodes / TILE_M;
        sage_gemm_ln<<<blocks, 256, 0, stream>>>(h, support, Wself, bself,
                                                 Wneigh, bneigh, gamma, beta,
                                                 (float*)d_out);
    }
}